// SSLGCN_59854664237657
// MI455X (gfx1250) — compile-verified
//
#include <hip/hip_runtime.h>

// ---------------------------------------------------------------------------
// Problem constants (match reference)
// ---------------------------------------------------------------------------
constexpr int kNUsers = 27094;
constexpr int kMSpots = 42852;
constexpr int kD      = 128;
constexpr int kEUS    = 600000;
constexpr int kESP    = 500000;

typedef __attribute__((ext_vector_type(2))) float v2f;
typedef __attribute__((ext_vector_type(8))) float v8f;

// ---------------------------------------------------------------------------
// Elementwise helpers
// ---------------------------------------------------------------------------
__global__ void k_copy(float* __restrict__ dst, const float* __restrict__ src, long long n) {
  long long i = (long long)blockIdx.x * blockDim.x + threadIdx.x;
  if (i < n) dst[i] = src[i];
}

__global__ void k_add(float* __restrict__ dst, const float* __restrict__ src, long long n) {
  long long i = (long long)blockIdx.x * blockDim.x + threadIdx.x;
  if (i < n) dst[i] += src[i];
}

__global__ void k_scale(float* __restrict__ dst, long long n, float a) {
  long long i = (long long)blockIdx.x * blockDim.x + threadIdx.x;
  if (i < n) dst[i] *= a;
}

// ---------------------------------------------------------------------------
// Degree / normalization kernels
// ---------------------------------------------------------------------------
__global__ void k_deg(const int* __restrict__ u, const int* __restrict__ s,
                      float* __restrict__ udeg, float* __restrict__ sdeg, int E) {
  int e = blockIdx.x * blockDim.x + threadIdx.x;
  if (e < E) {
    atomicAdd(&udeg[u[e]], 1.0f);
    atomicAdd(&sdeg[s[e]], 1.0f);
  }
}

__global__ void k_invdiv(const int* __restrict__ u, const int* __restrict__ s,
                         const float* __restrict__ udeg, const float* __restrict__ sdeg,
                         float* __restrict__ invd, int E) {
  int e = blockIdx.x * blockDim.x + threadIdx.x;
  if (e < E) invd[e] = 1.0f / sqrtf(udeg[u[e]] * sdeg[s[e]]);
}

__global__ void k_wdeg(const int* __restrict__ dst, const float* __restrict__ ew,
                       float* __restrict__ deg, int E) {
  int e = blockIdx.x * blockDim.x + threadIdx.x;
  if (e < E) atomicAdd(&deg[dst[e]], ew[e]);
}

__global__ void k_dinv(float* __restrict__ deg, int n) {
  int i = blockIdx.x * blockDim.x + threadIdx.x;
  if (i < n) {
    float d = deg[i];
    deg[i] = (d > 0.0f) ? (1.0f / sqrtf(fmaxf(d, 1e-12f))) : 0.0f;
  }
}

// ---------------------------------------------------------------------------
// Edge scatter: out[dst[e]] += w[e] * xsrc[src[e]]   (one wave per edge,
// each lane handles 4 of the 128 features via a float4 load + 4 f32 atomics)
// ---------------------------------------------------------------------------
__global__ void k_scatter_edge(const float* __restrict__ xsrc, const int* __restrict__ src,
                               const int* __restrict__ dst, const float* __restrict__ w,
                               float* __restrict__ out, int E) {
  long long t = (long long)blockIdx.x * blockDim.x + threadIdx.x;
  int e = (int)(t >> 5);
  if (e >= E) return;
  int lane = (int)(t & 31);
  float wv = w[e];
  float4 v = ((const float4*)(xsrc + (long long)src[e] * kD))[lane];
  float* dp = out + (long long)dst[e] * kD + lane * 4;
  atomicAdd(dp + 0, v.x * wv);
  atomicAdd(dp + 1, v.y * wv);
  atomicAdd(dp + 2, v.z * wv);
  atomicAdd(dp + 3, v.w * wv);
}

// GCN variant: weight = dinv[src] * ew * dinv[dst] * scale
__global__ void k_gcn_scatter(const float* __restrict__ xw, const int* __restrict__ src,
                              const int* __restrict__ dst, const float* __restrict__ ew,
                              const float* __restrict__ dinv, float* __restrict__ out,
                              float scale, int E) {
  long long t = (long long)blockIdx.x * blockDim.x + threadIdx.x;
  int e = (int)(t >> 5);
  if (e >= E) return;
  int lane = (int)(t & 31);
  int si = src[e], di = dst[e];
  float wv = dinv[si] * ew[e] * dinv[di] * scale;
  float4 v = ((const float4*)(xw + (long long)si * kD))[lane];
  float* dp = out + (long long)di * kD + lane * 4;
  atomicAdd(dp + 0, v.x * wv);
  atomicAdd(dp + 1, v.y * wv);
  atomicAdd(dp + 2, v.z * wv);
  atomicAdd(dp + 3, v.w * wv);
}

// ---------------------------------------------------------------------------
// Y = X @ W using V_WMMA_F32_16X16X4_F32.
// X: M x 128 row-major, W: 128 x 128 row-major, Y: M x 128.
// One wave computes one 16x16 output tile; K=128 walked in 32 WMMA steps.
// A layout (ISA 7.12.2, 32-bit A 16x4): lane = M row (lane&15),
//   VGPR0/1 hold K = {0,1} for lanes 0-15 and K = {2,3} for lanes 16-31.
// B layout (row striped across lanes): lane holds column N = lane&15,
//   same K split as A. C/D: VGPR r holds row r + 8*(lane>>4), col lane&15.
// ---------------------------------------------------------------------------
__global__ __launch_bounds__(128) void k_gemm_wmma(const float* __restrict__ X,
                                                   const float* __restrict__ W,
                                                   float* __restrict__ Y, int M) {
  const int lane = threadIdx.x & 31;
  const int wave = threadIdx.x >> 5;
  const int tile = blockIdx.x * 4 + wave;  // wave-uniform
  const int tm = tile >> 3;                // 128/16 = 8 N-tiles
  const int tn = tile & 7;
  if (tm * 16 >= M) return;                // wave-uniform exit: EXEC stays full for WMMA

  const int mrow   = tm * 16 + (lane & 15);
  const int mclamp = (mrow < M) ? mrow : (M - 1);
  const int kb     = (lane >> 4) << 1;     // 0 or 2
  const int ncol   = tn * 16 + (lane & 15);
  const float* __restrict__ xrow = X + (long long)mclamp * kD;

  v8f acc = {0.f, 0.f, 0.f, 0.f, 0.f, 0.f, 0.f, 0.f};
#pragma unroll
  for (int k0 = 0; k0 < kD; k0 += 4) {
    v2f a, b;
    a.x = xrow[k0 + kb];
    a.y = xrow[k0 + kb + 1];
    b.x = W[(k0 + kb) * kD + ncol];
    b.y = W[(k0 + kb + 1) * kD + ncol];
    acc = __builtin_amdgcn_wmma_f32_16x16x4_f32(false, a, false, b, (short)0, acc,
                                                false, false);
  }

  const int half = (lane >> 4) << 3;       // 0 or 8
#pragma unroll
  for (int r = 0; r < 8; ++r) {
    const int row = tm * 16 + half + r;
    if (row < M) Y[(long long)row * kD + ncol] = acc[r];
  }
}

// ---------------------------------------------------------------------------
// Host orchestration
// ---------------------------------------------------------------------------
static inline unsigned nblk(long long n, int b) { return (unsigned)((n + b - 1) / b); }

extern "C" void kernel_launch(void* const* d_in, const int* in_sizes, int n_in,
                              void* d_out, int out_size, void* d_ws, size_t ws_size,
                              hipStream_t stream) {
  (void)in_sizes; (void)n_in; (void)out_size; (void)ws_size;

  const float* spot_emb = (const float*)d_in[0];
  const float* user_emb = (const float*)d_in[1];
  const float* W1   = (const float*)d_in[2];
  const float* W10  = (const float*)d_in[3];
  const float* W20  = (const float*)d_in[4];
  const float* ew1  = (const float*)d_in[5];
  const float* ew10 = (const float*)d_in[6];
  const float* ew20 = (const float*)d_in[7];
  const int* us   = (const int*)d_in[8];
  const int* ei1  = (const int*)d_in[9];
  const int* ei10 = (const int*)d_in[10];
  const int* ei20 = (const int*)d_in[11];
  const int* uIdx = us;          // user_spot row 0
  const int* sIdx = us + kEUS;   // user_spot row 1

  const long long MD = (long long)kMSpots * kD;
  const long long ND = (long long)kNUsers * kD;

  // Workspace carve-up (1 KiB-aligned regions)
  float* base = (float*)d_ws;
  size_t off = 0;
  auto alloc = [&](size_t n) {
    float* p = base + off;
    off += (n + 255) & ~(size_t)255;
    return p;
  };
  float* udeg = alloc(kNUsers);
  float* sdeg = alloc(kMSpots);
  float* invd = alloc(kEUS);
  float* degg = alloc(kMSpots);
  float* xw   = alloc((size_t)MD);
  float* sxa  = alloc((size_t)MD);
  float* sxb  = alloc((size_t)MD);
  float* uxa  = alloc((size_t)ND);
  float* uxb  = alloc((size_t)ND);

  float* out = (float*)d_out;
  float* so1 = out;
  float* uo1 = so1 + MD;
  float* so2 = uo1 + ND;
  float* uo2 = so2 + MD;
  float* so3 = uo2 + ND;
  float* uo3 = so3 + MD;

  const int B = 256;
  const unsigned gE32   = nblk((long long)kEUS * 32, B);
  const unsigned gESP32 = nblk((long long)kESP * 32, B);
  const int gemmTiles = ((kMSpots + 15) / 16) * (kD / 16);
  const unsigned gGemm = nblk(gemmTiles, 4);

  // --- degrees & per-edge symmetric norm -----------------------------------
  hipMemsetAsync(udeg, 0, kNUsers * sizeof(float), stream);
  hipMemsetAsync(sdeg, 0, kMSpots * sizeof(float), stream);
  k_deg<<<nblk(kEUS, B), B, 0, stream>>>(uIdx, sIdx, udeg, sdeg, kEUS);
  k_invdiv<<<nblk(kEUS, B), B, 0, stream>>>(uIdx, sIdx, udeg, sdeg, invd, kEUS);

  // 3-layer bipartite propagation; acc buffers pre-seeded with layer-0 term.
  auto propagate = [&](float* s_acc, float* u_acc) -> float* {
    float* sc = sxa; float* sn = sxb;
    float* uc = uxa; float* un = uxb;
    for (int l = 0; l < 3; ++l) {
      hipMemsetAsync(un, 0, ND * sizeof(float), stream);
      hipMemsetAsync(sn, 0, MD * sizeof(float), stream);
      // user_new[u] += invd * spot_x[s] ; spot_new[s] += invd * user_x[u]
      k_scatter_edge<<<gE32, B, 0, stream>>>(sc, sIdx, uIdx, invd, un, kEUS);
      k_scatter_edge<<<gE32, B, 0, stream>>>(uc, uIdx, sIdx, invd, sn, kEUS);
      k_add<<<nblk(ND, B), B, 0, stream>>>(u_acc, un, ND);
      k_add<<<nblk(MD, B), B, 0, stream>>>(s_acc, sn, MD);
      float* t;
      t = sc; sc = sn; sn = t;
      t = uc; uc = un; un = t;
    }
    return sc;  // final spot_x
  };

  // GCN-light: target += scale * norm-scatter(xsrc @ Wm) over graph (ei, ew)
  auto gcn = [&](const float* xsrc, const float* Wm, const int* ei, const float* ew,
                 float* target, float scale) {
    k_gemm_wmma<<<gGemm, 128, 0, stream>>>(xsrc, Wm, xw, kMSpots);
    hipMemsetAsync(degg, 0, kMSpots * sizeof(float), stream);
    k_wdeg<<<nblk(kESP, B), B, 0, stream>>>(ei + kESP, ew, degg, kESP);
    k_dinv<<<nblk(kMSpots, B), B, 0, stream>>>(degg, kMSpots);
    k_gcn_scatter<<<gESP32, B, 0, stream>>>(xw, ei, ei + kESP, ew, degg, target,
                                            scale, kESP);
  };

  // --- shared propagation (branches 2 & 3) ---------------------------------
  k_copy<<<nblk(MD, B), B, 0, stream>>>(sxa, spot_emb, MD);
  k_copy<<<nblk(ND, B), B, 0, stream>>>(uxa, user_emb, ND);
  k_copy<<<nblk(MD, B), B, 0, stream>>>(so2, spot_emb, MD);
  k_copy<<<nblk(ND, B), B, 0, stream>>>(uo2, user_emb, ND);
  float* fin_s = propagate(so2, uo2);
  k_scale<<<nblk(MD, B), B, 0, stream>>>(so2, MD, 0.25f);
  k_scale<<<nblk(ND, B), B, 0, stream>>>(uo2, ND, 0.25f);
  k_copy<<<nblk(MD, B), B, 0, stream>>>(so3, so2, MD);
  k_copy<<<nblk(ND, B), B, 0, stream>>>(uo3, uo2, ND);

  // --- cat2 into so2, cat3 (from final spot_x) into so3 --------------------
  gcn(spot_emb, W10, ei10, ew10, so2, 0.25f);
  gcn(fin_s,    W20, ei20, ew20, so3, 0.25f);   // consumes fin_s before reuse

  // --- branch 1: x_s1 = spot_emb + cat1, then propagate --------------------
  k_copy<<<nblk(MD, B), B, 0, stream>>>(sxa, spot_emb, MD);
  gcn(spot_emb, W1, ei1, ew1, sxa, 1.0f);       // sxa = spot_emb + cat1
  k_copy<<<nblk(MD, B), B, 0, stream>>>(so1, sxa, MD);
  k_copy<<<nblk(ND, B), B, 0, stream>>>(uxa, user_emb, ND);
  k_copy<<<nblk(ND, B), B, 0, stream>>>(uo1, user_emb, ND);
  (void)propagate(so1, uo1);
  k_scale<<<nblk(MD, B), B, 0, stream>>>(so1, MD, 0.25f);
  k_scale<<<nblk(ND, B), B, 0, stream>>>(uo1, ND, 0.25f);
}